// KnowformerLayer_15951508537887
// MI455X (gfx1250) — compile-verified
//
#include <hip/hip_runtime.h>
#include <cstdint>

// ---------------- CDNA5 (gfx1250) Knowformer layer, f32 WMMA ----------------
// Memory-bound workload (rspmm atomics ~1.6GB dominate at 23.3 TB/s), so we
// keep full f32 precision and use V_WMMA_F32_16X16X4_F32 for all dense GEMMs.

typedef float v2f __attribute__((ext_vector_type(2)));
typedef float v8f __attribute__((ext_vector_type(8)));

constexpr int B_ = 4, V_ = 20000, R_ = 64, E_ = 200000, H_ = 4;
constexpr int NROW = B_ * V_;     // 80000 rows, divisible by 64
constexpr int STR = 65;           // LDS row stride (bank-conflict pad)
constexpr float LNEPS = 1e-5f;

__device__ __forceinline__ v8f wmma_f32(v2f a, v2f b, v8f c) {
  // D(16x16,f32) = A(16x4,f32) * B(4x16,f32) + C
  return __builtin_amdgcn_wmma_f32_16x16x4_f32(false, a, false, b, (short)0, c,
                                               false, false);
}

// ---------------------------------------------------------------------------
// Small projection: out[b,o] = z[b,:] . W[o,:] + bias[o]   (B x O, K=64)
__global__ void k_zproj(const float* __restrict__ z, const float* __restrict__ W,
                        const float* __restrict__ bias, float* __restrict__ out,
                        int O) {
  int i = blockIdx.x * blockDim.x + threadIdx.x;
  if (i >= B_ * O) return;
  int b = i / O, o = i - b * O;
  float s = bias[o];
#pragma unroll 8
  for (int k = 0; k < 64; ++k) s += z[b * 64 + k] * W[o * 64 + k];
  out[i] = s;
}

// ---------------------------------------------------------------------------
// rspmm: agg[b,dst,d] += zrel[b,r,d] * h[b,src,d]  (one block per edge)
__global__ __launch_bounds__(256)
void k_rspmm(const int* __restrict__ ei, const float* __restrict__ zrel,
             const float* __restrict__ h, float* __restrict__ agg) {
  int e = blockIdx.x;
  int t = threadIdx.x;
  int b = t >> 6, d = t & 63;
  int dst = ei[3 * e + 0];
  int r   = ei[3 * e + 1];
  int src = ei[3 * e + 2];
  float msg = zrel[(b * R_ + r) * 64 + d] * h[(b * V_ + src) * 64 + d];
  atomicAdd(&agg[(b * V_ + dst) * 64 + d], msg);
}

// ---------------------------------------------------------------------------
// Fused row MLP:  y = relu(t @ W1[:, :64]^T + b1 [+extra]) @ W2^T + b2
//   pre_mode  : 0 t=in            1 t=agg + alpha*in
//   extra_mode: 0 none            1 += noise[n]*W1[:,64]   2 one-hot rowsum
//   post_mode : 0 out=y           1 out=LN(y)*g+bl+in      2 out=LN(in+y)*g+bl
// Block = 128 threads (4 waves); each wave owns a 16-row x 64-col tile.
__global__ __launch_bounds__(128)
void k_mlp2(const float* in, const float* __restrict__ agg,
            const float* __restrict__ alpha,
            const float* __restrict__ W1, const float* __restrict__ b1, int w1c,
            const float* __restrict__ W2, const float* __restrict__ b2,
            const float* __restrict__ g, const float* __restrict__ bl,
            const float* __restrict__ extra, const int* __restrict__ hidx,
            float* out, int extra_mode, int pre_mode, int post_mode) {
  __shared__ float sW1[64 * STR];
  __shared__ float sW2[64 * STR];
  __shared__ float sT [64 * STR];
  __shared__ float sEx[64];
  __shared__ float sMu[64];
  __shared__ float sRs[64];

  const int tid  = threadIdx.x;
  const int wid  = tid >> 5;
  const int lane = tid & 31;
  const int half = lane >> 4;
  const int m    = lane & 15;
  const int lr0  = wid * 16;
  const int rowbase = blockIdx.x * 64;

  for (int i = tid; i < 64 * 64; i += 128) {
    int o = i >> 6, k = i & 63;
    sW1[o * STR + k] = W1[o * w1c + k];
    sW2[o * STR + k] = W2[o * 64 + k];
  }
  for (int i = tid; i < 64 * 64; i += 128) {
    int r = i >> 6, c = i & 63;
    int n = rowbase + r;
    float t = in[n * 64 + c];
    if (pre_mode == 1) t = agg[n * 64 + c] + alpha[c] * t;
    sT[r * STR + c] = t;
  }
  if (extra_mode == 2 && tid < 64) {
    float s = 0.f;
    for (int k = 64; k < 128; ++k) s += W1[tid * w1c + k];
    sEx[tid] = s;
  }
  __syncthreads();

  // ---- GEMM1: y1 = t @ W1^T (K=64, 16 k-steps, 4 col tiles) ----
  v8f acc[4];
#pragma unroll
  for (int ct = 0; ct < 4; ++ct) acc[ct] = (v8f){0,0,0,0,0,0,0,0};
#pragma unroll
  for (int k0 = 0; k0 < 64; k0 += 4) {
    v2f a;
    a.x = sT[(lr0 + m) * STR + k0 + 2 * half];
    a.y = sT[(lr0 + m) * STR + k0 + 2 * half + 1];
#pragma unroll
    for (int ct = 0; ct < 4; ++ct) {
      int o = ct * 16 + m;
      v2f bb;
      bb.x = sW1[o * STR + k0 + 2 * half];
      bb.y = sW1[o * STR + k0 + 2 * half + 1];
      acc[ct] = wmma_f32(a, bb, acc[ct]);
    }
  }

  // bias + extras + relu -> y1 back into sT (wave-private 16-row region)
#pragma unroll
  for (int ct = 0; ct < 4; ++ct) {
    int o = ct * 16 + m;
    float bz  = b1[o];
    float wex = (extra_mode == 1) ? W1[o * w1c + 64] : 0.f;
#pragma unroll
    for (int j = 0; j < 8; ++j) {
      int r = lr0 + j + 8 * half;
      int n = rowbase + r;
      float y = acc[ct][j] + bz;
      if (extra_mode == 1) y += extra[n] * wex;
      if (extra_mode == 2) {
        int b = n / V_;
        if ((n - b * V_) == hidx[b]) y += sEx[o];
      }
      sT[r * STR + o] = fmaxf(y, 0.f);
    }
  }

  // ---- GEMM2: y2 = y1 @ W2^T ----
  v8f acc2[4];
#pragma unroll
  for (int ct = 0; ct < 4; ++ct) acc2[ct] = (v8f){0,0,0,0,0,0,0,0};
#pragma unroll
  for (int k0 = 0; k0 < 64; k0 += 4) {
    v2f a;
    a.x = sT[(lr0 + m) * STR + k0 + 2 * half];
    a.y = sT[(lr0 + m) * STR + k0 + 2 * half + 1];
#pragma unroll
    for (int ct = 0; ct < 4; ++ct) {
      int o = ct * 16 + m;
      v2f bb;
      bb.x = sW2[o * STR + k0 + 2 * half];
      bb.y = sW2[o * STR + k0 + 2 * half + 1];
      acc2[ct] = wmma_f32(a, bb, acc2[ct]);
    }
  }

  if (post_mode == 0) {
#pragma unroll
    for (int ct = 0; ct < 4; ++ct) {
      int o = ct * 16 + m;
      float bz = b2[o];
#pragma unroll
      for (int j = 0; j < 8; ++j) {
        int r = lr0 + j + 8 * half;
        out[(rowbase + r) * 64 + o] = acc2[ct][j] + bz;
      }
    }
    return;
  }

  // stash y2 (+res for post 2) then LayerNorm per row
#pragma unroll
  for (int ct = 0; ct < 4; ++ct) {
    int o = ct * 16 + m;
    float bz = b2[o];
#pragma unroll
    for (int j = 0; j < 8; ++j) {
      int r = lr0 + j + 8 * half;
      int n = rowbase + r;
      float y = acc2[ct][j] + bz;
      if (post_mode == 2) y += in[n * 64 + o];
      sT[r * STR + o] = y;
    }
  }
  if (lane < 16) {
    int r = lr0 + lane;
    float s = 0.f;
    for (int c = 0; c < 64; ++c) s += sT[r * STR + c];
    float mu = s * (1.f / 64.f), vv = 0.f;
    for (int c = 0; c < 64; ++c) { float d = sT[r * STR + c] - mu; vv += d * d; }
    sMu[r] = mu;
    sRs[r] = rsqrtf(vv * (1.f / 64.f) + LNEPS);
  }
  for (int i = lane; i < 16 * 64; i += 32) {
    int r = lr0 + (i >> 6), c = i & 63;
    int n = rowbase + r;
    float y = (sT[r * STR + c] - sMu[r]) * sRs[r] * g[c] + bl[c];
    if (post_mode == 1) y += in[n * 64 + c];
    out[n * 64 + c] = y;
  }
}

// ---------------------------------------------------------------------------
// qk = qk_x @ W^T + b  with W (128,64): 16-row x 128-col tile per wave
__global__ __launch_bounds__(128)
void k_linear128(const float* __restrict__ in, const float* __restrict__ W,
                 const float* __restrict__ bias, float* __restrict__ out) {
  __shared__ float sW[128 * STR];
  __shared__ float sT[64 * STR];
  const int tid  = threadIdx.x;
  const int wid  = tid >> 5, lane = tid & 31;
  const int half = lane >> 4, m = lane & 15;
  const int lr0  = wid * 16;
  const int rowbase = blockIdx.x * 64;

  for (int i = tid; i < 128 * 64; i += 128) {
    int o = i >> 6, k = i & 63;
    sW[o * STR + k] = W[i];
  }
  for (int i = tid; i < 64 * 64; i += 128) {
    int r = i >> 6, c = i & 63;
    sT[r * STR + c] = in[(rowbase + r) * 64 + c];
  }
  __syncthreads();

  for (int ct = 0; ct < 8; ++ct) {
    v8f acc = (v8f){0,0,0,0,0,0,0,0};
    int o = ct * 16 + m;
#pragma unroll
    for (int k0 = 0; k0 < 64; k0 += 4) {
      v2f a, bb;
      a.x  = sT[(lr0 + m) * STR + k0 + 2 * half];
      a.y  = sT[(lr0 + m) * STR + k0 + 2 * half + 1];
      bb.x = sW[o * STR + k0 + 2 * half];
      bb.y = sW[o * STR + k0 + 2 * half + 1];
      acc = wmma_f32(a, bb, acc);
    }
    float bz = bias[o];
#pragma unroll
    for (int j = 0; j < 8; ++j) {
      int r = lr0 + j + 8 * half;
      out[(rowbase + r) * 128 + o] = acc[j] + bz;
    }
  }
}

// ---------------------------------------------------------------------------
// Linear-attention reduction: per (b,h): kvs = k_n^T v (WMMA over rows),
// ksum = sum k_n, vsum = sum v.  stats[g*288]: [0,256)=kvs, +256 ksum, +272 vsum
constexpr int CW = 160, NCH = V_ / CW;   // 125 chunks
__global__ __launch_bounds__(64)
void k_attn_red(const float* __restrict__ qk, const float* __restrict__ vx,
                float* __restrict__ stats) {
  __shared__ float skn[64 * 17];
  __shared__ float svv[64 * 17];
  const int chunk = blockIdx.x % NCH;
  const int gidx  = blockIdx.x / NCH;       // b*H + h
  const int b = gidx / H_, h = gidx % H_;
  const int tid = threadIdx.x, wid = tid >> 5, lane = tid & 31;
  const int half = lane >> 4, m = lane & 15;

  v8f acc = (v8f){0,0,0,0,0,0,0,0};
  float red = 0.f;

  for (int t0 = 0; t0 < CW; t0 += 64) {
    int v = chunk * CW + t0 + tid;
    bool ok = (t0 + tid < CW) && (v < V_);
    {
      float kr[16]; float nr = 0.f;
      int kb = (b * V_ + v) * 128 + 64 + h * 16;
#pragma unroll
      for (int j = 0; j < 16; ++j) { float x = ok ? qk[kb + j] : 0.f; kr[j] = x; nr += x * x; }
      float inv = 1.f / fmaxf(sqrtf(nr), 1e-12f);
#pragma unroll
      for (int j = 0; j < 16; ++j) skn[tid * 17 + j] = ok ? kr[j] * inv : 0.f;
      int vb = (b * V_ + v) * 64 + h * 16;
#pragma unroll
      for (int j = 0; j < 16; ++j) svv[tid * 17 + j] = ok ? vx[vb + j] : 0.f;
    }
    __syncthreads();
    if (wid == 0) {
#pragma unroll
      for (int k0 = 0; k0 < 64; k0 += 4) {   // kvs[d][D] += kn[row][d]*vv[row][D]
        v2f a, bb;
        a.x  = skn[(k0 + 2 * half) * 17 + m];
        a.y  = skn[(k0 + 2 * half + 1) * 17 + m];
        bb.x = svv[(k0 + 2 * half) * 17 + m];
        bb.y = svv[(k0 + 2 * half + 1) * 17 + m];
        acc = wmma_f32(a, bb, acc);
      }
    } else {
      if (lane < 16) { for (int r = 0; r < 64; ++r) red += skn[r * 17 + lane]; }
      else           { for (int r = 0; r < 64; ++r) red += svv[r * 17 + (lane - 16)]; }
    }
    __syncthreads();
  }
  float* gs = stats + gidx * 288;
  if (wid == 0) {
#pragma unroll
    for (int j = 0; j < 8; ++j) atomicAdd(&gs[(j + 8 * half) * 16 + m], acc[j]);
  } else {
    atomicAdd(&gs[256 + lane], red);   // 256..271 ksum, 272..287 vsum
  }
}

// per (b,v,h): out = (q_n@kvs + vsum + v*n) / (q_n.ksum + 2n)
__global__ __launch_bounds__(256)
void k_attn_apply(const float* __restrict__ qk, const float* __restrict__ vx,
                  const float* __restrict__ stats, float* __restrict__ out) {
  __shared__ float ss[16 * 288];
  for (int i = threadIdx.x; i < 16 * 288; i += 256) ss[i] = stats[i];
  __syncthreads();
  int idx = blockIdx.x * 256 + threadIdx.x;
  if (idx >= B_ * V_ * H_) return;
  int h = idx % H_, bv = idx / H_;
  const float* gs = ss + ((bv / V_) * H_ + h) * 288;
  float qn[16]; float nr = 0.f;
  int qb = bv * 128 + h * 16;
#pragma unroll
  for (int j = 0; j < 16; ++j) { float x = qk[qb + j]; qn[j] = x; nr += x * x; }
  float inv = 1.f / fmaxf(sqrtf(nr), 1e-12f);
  float den = 2.f * (float)V_;
#pragma unroll
  for (int j = 0; j < 16; ++j) { qn[j] *= inv; den += qn[j] * gs[256 + j]; }
  int vb = bv * 64 + h * 16;
  float rden = 1.f / den;
#pragma unroll
  for (int l = 0; l < 16; ++l) {
    float num = gs[272 + l] + vx[vb + l] * (float)V_;
#pragma unroll
    for (int j = 0; j < 16; ++j) num += qn[j] * gs[j * 16 + l];
    out[vb + l] = num * rden;
  }
}

// out = LN(a + d)*g + b   (4 rows per 256-thread block)
__global__ __launch_bounds__(256)
void k_add_ln(const float* __restrict__ a, const float* __restrict__ d,
              const float* __restrict__ g, const float* __restrict__ bb,
              float* __restrict__ out) {
  __shared__ float s[4 * 64];
  __shared__ float mu[4], rs[4];
  int tid = threadIdx.x, rl = tid >> 6, c = tid & 63;
  int n = blockIdx.x * 4 + rl;
  float x = a[n * 64 + c] + d[n * 64 + c];
  s[rl * 64 + c] = x;
  __syncthreads();
  if (c == 0) {
    float sum = 0.f;
    for (int k = 0; k < 64; ++k) sum += s[rl * 64 + k];
    float m_ = sum * (1.f / 64.f), vv = 0.f;
    for (int k = 0; k < 64; ++k) { float dd = s[rl * 64 + k] - m_; vv += dd * dd; }
    mu[rl] = m_; rs[rl] = rsqrtf(vv * (1.f / 64.f) + LNEPS);
  }
  __syncthreads();
  out[n * 64 + c] = (x - mu[rl]) * rs[rl] * g[c] + bb[c];
}

// ---------------------------------------------------------------------------
extern "C" void kernel_launch(void* const* d_in, const int* in_sizes, int n_in,
                              void* d_out, int out_size, void* d_ws, size_t ws_size,
                              hipStream_t stream) {
  (void)in_sizes; (void)n_in; (void)out_size; (void)ws_size;
  const float* x        = (const float*)d_in[0];
  const float* z        = (const float*)d_in[1];
  const float* noise    = (const float*)d_in[3];
  const float* fqz_w    = (const float*)d_in[4];
  const float* fqz_b    = (const float*)d_in[5];
  const float* fqx_w1   = (const float*)d_in[6];
  const float* fqx_b1   = (const float*)d_in[7];
  const float* fqx_w2   = (const float*)d_in[8];
  const float* fqx_b2   = (const float*)d_in[9];
  const float* fvx_w1   = (const float*)d_in[10];
  const float* fvx_b1   = (const float*)d_in[11];
  const float* fvx_w2   = (const float*)d_in[12];
  const float* fvx_b2   = (const float*)d_in[13];
  const float* ftq_w    = (const float*)d_in[14];
  const float* ftq_b    = (const float*)d_in[15];
  const float* qk_w1    = (const float*)d_in[16];
  const float* qk_b1    = (const float*)d_in[17];
  const float* qk_w2    = (const float*)d_in[18];
  const float* qk_b2    = (const float*)d_in[19];
  const float* qk_alpha = (const float*)d_in[20];
  const float* qk_ng    = (const float*)d_in[21];
  const float* qk_nb    = (const float*)d_in[22];
  const float* v_fcz_w  = (const float*)d_in[23];
  const float* v_fcz_b  = (const float*)d_in[24];
  const float* v_w1     = (const float*)d_in[25];
  const float* v_b1     = (const float*)d_in[26];
  const float* v_w2     = (const float*)d_in[27];
  const float* v_b2     = (const float*)d_in[28];
  const float* v_beta   = (const float*)d_in[29];
  const float* v_ng     = (const float*)d_in[30];
  const float* v_nb     = (const float*)d_in[31];
  const float* ffn_w1   = (const float*)d_in[32];
  const float* ffn_b1   = (const float*)d_in[33];
  const float* ffn_w2   = (const float*)d_in[34];
  const float* ffn_b2   = (const float*)d_in[35];
  const float* attn_g   = (const float*)d_in[36];
  const float* attn_b   = (const float*)d_in[37];
  const float* norm_g   = (const float*)d_in[38];
  const float* norm_b   = (const float*)d_in[39];
  const int*   h_index  = (const int*)d_in[40];
  const int*   edge_idx = (const int*)d_in[41];

  // workspace carve-up (floats)
  float* ws   = (float*)d_ws;
  float* qk_z = ws;                              // B*R*64        = 16384
  float* vz   = qk_z + B_ * R_ * 64;             // 2 * 16384
  float* qk_x = vz + 2 * B_ * R_ * 64;           // NROW*64
  float* v_x  = qk_x + NROW * 64;                // NROW*64
  float* aggb = v_x + NROW * 64;                 // NROW*64 (also attn_raw)
  float* qkb  = aggb + NROW * 64;                // NROW*128
  float* stat = qkb + NROW * 128;                // 16*288
  float* x1   = qk_x;                            // reuse after k_linear128

  const int GZ = (B_ * R_ * 64 + 255) / 256;
  k_zproj<<<GZ, 256, 0, stream>>>(z, fqz_w, fqz_b, qk_z, R_ * 64);
  for (int i = 0; i < 2; ++i)
    k_zproj<<<GZ, 256, 0, stream>>>(z, v_fcz_w + i * R_ * 64 * 64,
                                    v_fcz_b + i * R_ * 64, vz + i * B_ * R_ * 64,
                                    R_ * 64);

  const int GB = NROW / 64;  // 1250
  // qk_x = mlp2(concat(x, noise))
  k_mlp2<<<GB, 128, 0, stream>>>(x, nullptr, nullptr, fqx_w1, fqx_b1, 65,
                                 fqx_w2, fqx_b2, nullptr, nullptr, noise, nullptr,
                                 qk_x, 1, 0, 0);
  for (int i = 0; i < 2; ++i) {
    hipMemsetAsync(aggb, 0, (size_t)NROW * 64 * sizeof(float), stream);
    k_rspmm<<<E_, 256, 0, stream>>>(edge_idx, qk_z, qk_x, aggb);
    k_mlp2<<<GB, 128, 0, stream>>>(qk_x, aggb, qk_alpha + i * 64,
                                   qk_w1 + i * 4096, qk_b1 + i * 64, 64,
                                   qk_w2 + i * 4096, qk_b2 + i * 64,
                                   qk_ng + i * 64, qk_nb + i * 64,
                                   nullptr, nullptr, qk_x, 0, 1, 1);
  }

  // v_x = mlp2(concat(x, one-hot))
  k_mlp2<<<GB, 128, 0, stream>>>(x, nullptr, nullptr, fvx_w1, fvx_b1, 128,
                                 fvx_w2, fvx_b2, nullptr, nullptr, nullptr,
                                 h_index, v_x, 2, 0, 0);
  for (int i = 0; i < 2; ++i) {
    hipMemsetAsync(aggb, 0, (size_t)NROW * 64 * sizeof(float), stream);
    k_rspmm<<<E_, 256, 0, stream>>>(edge_idx, vz + i * B_ * R_ * 64, v_x, aggb);
    k_mlp2<<<GB, 128, 0, stream>>>(v_x, aggb, v_beta + i * 64,
                                   v_w1 + i * 4096, v_b1 + i * 64, 64,
                                   v_w2 + i * 4096, v_b2 + i * 64,
                                   v_ng + i * 64, v_nb + i * 64,
                                   nullptr, nullptr, v_x, 0, 1, 1);
  }

  // qk = qk_x @ fc_to_qk^T + b
  k_linear128<<<GB, 128, 0, stream>>>(qk_x, ftq_w, ftq_b, qkb);

  // linear attention
  hipMemsetAsync(stat, 0, 16 * 288 * sizeof(float), stream);
  k_attn_red<<<B_ * H_ * NCH, 64, 0, stream>>>(qkb, v_x, stat);
  k_attn_apply<<<(B_ * V_ * H_ + 255) / 256, 256, 0, stream>>>(qkb, v_x, stat, aggb);
  k_add_ln<<<NROW / 4, 256, 0, stream>>>(x, aggb, attn_g, attn_b, x1);

  // final: out = LN(x1 + mlp2(x1))
  k_mlp2<<<GB, 128, 0, stream>>>(x1, nullptr, nullptr, ffn_w1, ffn_b1, 64,
                                 ffn_w2, ffn_b2, norm_g, norm_b, nullptr, nullptr,
                                 (float*)d_out, 0, 0, 2);
}